// VolumetricFeatureSampler_20298015441560
// MI455X (gfx1250) — compile-verified
//
#include <hip/hip_runtime.h>
#include <hip/hip_bf16.h>

// Problem constants from the reference setup
#define BB 2
#define VV 4
#define FF 32
#define HH 256
#define WW 256
#define NPTS (64 * 64 * 64)   // GH*GW*GD, G=1
#define HWSZ (HH * WW)

typedef __attribute__((ext_vector_type(2))) float v2f;
typedef __attribute__((ext_vector_type(8))) float v8f;

// ---------------------------------------------------------------------------
// KRT = Ks (3x3) @ RTs (3x4), done on the matrix core.
// One 32-lane block per (b,v). A = Ks padded to 16x4 (col 3 = 0, rows 3..15 = 0),
// B = RTs padded to 4x16 (row 3 = 0, cols 4..15 = 0).
// A layout (f32 16x4): lanes 0-15 hold row m: V0=K0, V1=K1; lanes 16-31: V0=K2, V1=K3.
// B layout (f32 4x16): lanes 0-15 hold col n: V0=K0, V1=K1; lanes 16-31: V0=K2, V1=K3.
// D layout (f32 16x16): VGPR r, lanes 0-15 = row r, col = lane.
// ---------------------------------------------------------------------------
__global__ __launch_bounds__(32) void krt_wmma_kernel(const float* __restrict__ Ks,
                                                      const float* __restrict__ RTs,
                                                      float* __restrict__ KRT) {
    const int bv = blockIdx.x;                 // 0 .. B*V-1
    const float* K = Ks + bv * 9;              // 3x3 row-major
    const float* R = RTs + bv * 12;            // 3x4 row-major

    const int lane = threadIdx.x;
    const int r = lane & 15;
    const bool hi = lane >= 16;

    // Branchless operand construction (clamped indices keep loads in-bounds).
    const int rc = (r < 3) ? r : 0;            // A row clamp (only rows 0..2 real)
    float a0 = hi ? K[rc * 3 + 2] : K[rc * 3 + 0];
    float a1 = hi ? 0.0f          : K[rc * 3 + 1];
    a0 = (r < 3) ? a0 : 0.0f;
    a1 = (r < 3) ? a1 : 0.0f;

    const int nc = (r < 4) ? r : 0;            // B col clamp (only cols 0..3 real)
    float b0 = hi ? R[2 * 4 + nc] : R[0 * 4 + nc];
    float b1 = hi ? 0.0f          : R[1 * 4 + nc];
    b0 = (r < 4) ? b0 : 0.0f;
    b1 = (r < 4) ? b1 : 0.0f;

    v2f a; a[0] = a0; a[1] = a1;
    v2f b; b[0] = b0; b[1] = b1;
    v8f c = {};

    // D = A x B + 0   (EXEC is all-1s here: full 32-lane block, no divergence)
    v8f d = __builtin_amdgcn_wmma_f32_16x16x4_f32(
        /*neg_a=*/false, a, /*neg_b=*/false, b,
        /*c_mod=*/(short)0, c, /*reuse_a=*/false, /*reuse_b=*/false);

    // KRT[bv][i][j] = D row i (VGPR i), col j (lane j), i<3, j<4.
    if (lane < 4) {
        float* o = KRT + bv * 12;
        o[0 * 4 + lane] = d[0];
        o[1 * 4 + lane] = d[1];
        o[2 * 4 + lane] = d[2];
    }
}

// ---------------------------------------------------------------------------
// Main gather kernel: one thread per grid point. Hoists per-view projection
// into 4 weights (validity folded in) + 4 clamped offsets, then streams the
// 32 channels, writing mean and variance directly (no second pass).
// ---------------------------------------------------------------------------
__global__ __launch_bounds__(256) void vfs_sample_kernel(const float* __restrict__ feat,
                                                         const float* __restrict__ grids,
                                                         const float* __restrict__ KRT,
                                                         float* __restrict__ out) {
    const int b = blockIdx.y;
    const int n = blockIdx.x * 256 + threadIdx.x;   // point index within batch

    // grids: (B, 1, 3, 64,64,64) -> axis-major per batch
    const float gx = grids[((size_t)b * 3 + 0) * NPTS + n];
    const float gy = grids[((size_t)b * 3 + 1) * NPTS + n];
    const float gz = grids[((size_t)b * 3 + 2) * NPTS + n];

    float wgt[VV][4];
    int   off[VV][4];

    const float* kr = KRT + (size_t)b * VV * 12;

#pragma unroll
    for (int vw = 0; vw < VV; ++vw) {
        const float* k = kr + vw * 12;
        const float u0 = k[0] * gx + k[1] * gy + k[2]  * gz + k[3];
        const float v0 = k[4] * gx + k[5] * gy + k[6]  * gz + k[7];
        const float d0 = k[8] * gx + k[9] * gy + k[10] * gz + k[11];

        // Match reference math ordering exactly
        const float uu = 2.0f * u0 / d0 / (WW - 1.0f) - 1.0f;
        const float vv = 2.0f * v0 / d0 / (HH - 1.0f) - 1.0f;
        const float px = ((uu + 1.0f) * WW - 1.0f) * 0.5f;
        const float py = ((vv + 1.0f) * HH - 1.0f) * 0.5f;

        const float x0 = floorf(px), y0 = floorf(py);
        const float wx1 = px - x0, wx0 = 1.0f - wx1;
        const float wy1 = py - y0, wy0 = 1.0f - wy1;

        const float cx[2]  = {x0, x0 + 1.0f};
        const float cy[2]  = {y0, y0 + 1.0f};
        const float cwx[2] = {wx0, wx1};
        const float cwy[2] = {wy0, wy1};

#pragma unroll
        for (int cj = 0; cj < 2; ++cj) {
#pragma unroll
            for (int ci = 0; ci < 2; ++ci) {
                const float xi = cx[ci], yi = cy[cj];
                const bool valid = (xi >= 0.0f) & (xi <= WW - 1.0f) &
                                   (yi >= 0.0f) & (yi <= HH - 1.0f);
                const int xc = (int)fminf(fmaxf(xi, 0.0f), WW - 1.0f);
                const int yc = (int)fminf(fmaxf(yi, 0.0f), HH - 1.0f);
                off[vw][cj * 2 + ci] = yc * WW + xc;                       // always in-bounds
                wgt[vw][cj * 2 + ci] = cwx[ci] * cwy[cj] * (valid ? 1.0f : 0.0f);
            }
        }
    }

    const float* fb = feat + (size_t)b * VV * FF * HWSZ;
    float* ob = out + (size_t)b * 2 * FF * NPTS + n;

#pragma unroll 4
    for (int f = 0; f < FF; ++f) {
        float mean = 0.0f, msq = 0.0f;
#pragma unroll
        for (int vw = 0; vw < VV; ++vw) {
            const float* p = fb + ((size_t)vw * FF + f) * HWSZ;
            float s = wgt[vw][0] * p[off[vw][0]];
            s = fmaf(wgt[vw][1], p[off[vw][1]], s);
            s = fmaf(wgt[vw][2], p[off[vw][2]], s);
            s = fmaf(wgt[vw][3], p[off[vw][3]], s);
            mean += s;
            msq = fmaf(s, s, msq);
        }
        mean *= 0.25f;                       // 1/V
        msq  *= 0.25f;
        ob[(size_t)f * NPTS]        = mean;            // channels [0,F): mean
        ob[(size_t)(FF + f) * NPTS] = msq - mean * mean; // channels [F,2F): var
    }
}

extern "C" void kernel_launch(void* const* d_in, const int* in_sizes, int n_in,
                              void* d_out, int out_size, void* d_ws, size_t ws_size,
                              hipStream_t stream) {
    const float* feat  = (const float*)d_in[0];  // (B,V,F,H,W)
    const float* RTs   = (const float*)d_in[1];  // (B,V,3,4)
    const float* Ks    = (const float*)d_in[2];  // (B,V,3,3)
    const float* grids = (const float*)d_in[3];  // (B,1,3,64,64,64)
    float* out = (float*)d_out;
    float* krt = (float*)d_ws;                   // B*V*12 floats

    // Stage 1: KRT = Ks @ RTs via matrix core (one WMMA per view matrix)
    krt_wmma_kernel<<<dim3(BB * VV), dim3(32), 0, stream>>>(Ks, RTs, krt);

    // Stage 2: projection + bilinear gather + mean/var fusion
    dim3 grid(NPTS / 256, BB);
    vfs_sample_kernel<<<grid, dim3(256), 0, stream>>>(feat, grids, krt, out);
}